// SparseEncoderList_37374805410623
// MI455X (gfx1250) — compile-verified
//
#include <hip/hip_runtime.h>

// CTR-style per-field embedding gather for MI455X (gfx1250).
//   idx : [26, 16384] int32
//   W   : [26, 100000, 64] float32
//   out : [16384, 64, 26] float32, out[b,e,f] = W[f, idx[f,b], e]
//
// Pure bandwidth problem (~218 MB total @ 23.3 TB/s => ~9.4 us floor).
// Strategy: wave32 gathers 26 embedding rows straight into LDS via the
// CDNA5 async global->LDS path (b128, ASYNCcnt), then writes the
// transposed [64][26] tile back with contiguous b128 stores.

#define N_FIELDS 26
#define VOCAB    100000
#define EMB      64
#define BATCH    16384
#define TILE     (N_FIELDS * EMB)   // 1664 floats per sample
#define WAVES    8
#define BLOCK    (WAVES * 32)

#if defined(__has_builtin)
#  if __has_builtin(__builtin_amdgcn_global_load_async_to_lds_b128)
#    define HAVE_ASYNC_LDS 1
#  endif
#endif

// Builtin expects pointers to 16-byte int vectors (per hipcc diagnostic),
// in global (AS1) / LDS (AS3) address spaces.
typedef int v4i __attribute__((vector_size(16)));
typedef __attribute__((address_space(1))) v4i GlobalV4;
typedef __attribute__((address_space(3))) v4i LdsV4;

__global__ __launch_bounds__(BLOCK)
void ctr_embed_gather_kernel(const int* __restrict__ idx,
                             const float* __restrict__ W,
                             float* __restrict__ out)
{
    __shared__ float tile[WAVES * TILE];          // 53,248 B of 320 KB/WGP

    const int lane = threadIdx.x & 31;
    const int wave = threadIdx.x >> 5;
    const int b    = blockIdx.x * WAVES + wave;   // sample owned by this wave

    float* t = &tile[wave * TILE];                // wave-private [f][64] tile

    // ---- Phase 1: gather 26 rows (256 B each) into LDS, layout [f][64].
    // Half-wave per row: lanes 0..15 -> field 2i, lanes 16..31 -> field 2i+1.
    const int chunk = lane & 15;                  // 16-byte chunk within row
    const int fh    = lane >> 4;                  // which row of the pair
#pragma unroll
    for (int i = 0; i < N_FIELDS / 2; ++i) {
        const int f   = 2 * i + fh;
        const int row = idx[f * BATCH + b];
        const float* src = W + ((size_t)f * VOCAB + (size_t)row) * EMB + chunk * 4;
        float*       dst = t + f * EMB + chunk * 4;
#if HAVE_ASYNC_LDS
        // gfx1250 async VMEM->LDS gather, 16 B/lane, no VGPR round-trip.
        __builtin_amdgcn_global_load_async_to_lds_b128(
            (GlobalV4*)(void*)src, (LdsV4*)(void*)dst, 0, 0);
#else
        *(float4*)dst = *(const float4*)src;      // global_load_b128 + ds_store_b128
#endif
    }

#if HAVE_ASYNC_LDS
#  if __has_builtin(__builtin_amdgcn_s_wait_asynccnt)
    __builtin_amdgcn_s_wait_asynccnt(0);
#  else
    asm volatile("s_wait_asynccnt 0" ::: "memory");
#  endif
#endif
    __syncthreads();

    // ---- Phase 2: transposed write-out. Sample tile in `out` is 1664
    // contiguous floats: out[b*1664 + e*26 + f] = t[f*64 + e].
    // 13 iterations x 32 lanes x float4 = full-width b128 stores.
    const size_t outBase = (size_t)b * TILE;
#pragma unroll
    for (int i = 0; i < TILE / (32 * 4); ++i) {
        const int k0 = (i * 32 + lane) * 4;
        float4 v;
        {
            const int k = k0 + 0; const int e = k / N_FIELDS; const int f = k - e * N_FIELDS;
            v.x = t[f * EMB + e];
        }
        {
            const int k = k0 + 1; const int e = k / N_FIELDS; const int f = k - e * N_FIELDS;
            v.y = t[f * EMB + e];
        }
        {
            const int k = k0 + 2; const int e = k / N_FIELDS; const int f = k - e * N_FIELDS;
            v.z = t[f * EMB + e];
        }
        {
            const int k = k0 + 3; const int e = k / N_FIELDS; const int f = k - e * N_FIELDS;
            v.w = t[f * EMB + e];
        }
        *(float4*)(out + outBase + k0) = v;
    }
}

extern "C" void kernel_launch(void* const* d_in, const int* in_sizes, int n_in,
                              void* d_out, int out_size, void* d_ws, size_t ws_size,
                              hipStream_t stream)
{
    (void)in_sizes; (void)n_in; (void)out_size; (void)d_ws; (void)ws_size;
    const int*   idx = (const int*)d_in[0];      // [26, 16384] int32
    const float* W   = (const float*)d_in[1];    // [26, 100000, 64] float32
    float*       out = (float*)d_out;            // [16384, 64, 26] float32

    dim3 grid(BATCH / WAVES);                    // 2048 blocks
    dim3 block(BLOCK);                           // 256 threads = 8 wave32
    ctr_embed_gather_kernel<<<grid, block, 0, stream>>>(idx, W, out);
}